// MaritimeGATEncoder_14577119003283
// MI455X (gfx1250) — compile-verified
//
#include <hip/hip_runtime.h>
#include <hip/hip_bf16.h>

// MaritimeGATEncoder for MI455X (gfx1250, wave32).
// Heavy op: attention-weighted aggregation -> V_WMMA_F32_16X16X4_F32,
// attention weights computed per-lane directly in the WMMA A-operand layout.

#define GN   4096   // nodes
#define GH   4      // heads layer 1
#define GF1  64     // per-head dim layer 1
#define GC1  256    // H*F1
#define GF2  32     // layer 2 dim

typedef __attribute__((ext_vector_type(2))) float v2f;
typedef __attribute__((ext_vector_type(8))) float v8f;

__device__ __forceinline__ float lrelu02(float x) { return x > 0.f ? x : 0.2f * x; }

// ---------------------------------------------------------------------------
// Layer-1 projections: proj = x@w1, skipp = x@skip1   (K=5, VALU-bound)
// one thread per (n, c) of [N, 256]
// ---------------------------------------------------------------------------
__global__ void __launch_bounds__(256)
k_l1_proj(const float* __restrict__ x, const float* __restrict__ w1,
          const float* __restrict__ skip1,
          float* __restrict__ proj, float* __restrict__ skipp)
{
  int idx = blockIdx.x * 256 + threadIdx.x;      // < N*256 exactly
  int n = idx >> 8, c = idx & (GC1 - 1);
  const float* xr = x + n * 5;
  float p = 0.f, s = 0.f;
#pragma unroll
  for (int k = 0; k < 5; ++k) {
    float xv = xr[k];
    p = fmaf(xv, w1[k * GC1 + c], p);
    s = fmaf(xv, skip1[k * GC1 + c], s);
  }
  proj[idx] = p;
  skipp[idx] = s;
}

// ---------------------------------------------------------------------------
// Layer-1 attention scalars: s_src/s_tgt stored transposed [H, N]
// one thread per (h, n)
// ---------------------------------------------------------------------------
__global__ void __launch_bounds__(256)
k_l1_scores(const float* __restrict__ proj, const float* __restrict__ asrc,
            const float* __restrict__ atgt,
            float* __restrict__ ssrc, float* __restrict__ stgt)
{
  int idx = blockIdx.x * 256 + threadIdx.x;      // h*N + n, < GH*GN
  int h = idx >> 12, n = idx & (GN - 1);
  const float* pr = proj + n * GC1 + h * GF1;
  const float* as = asrc + h * GF1;
  const float* at = atgt + h * GF1;
  float s0 = 0.f, s1 = 0.f;
#pragma unroll 8
  for (int f = 0; f < GF1; ++f) {
    float pv = pr[f];
    s0 = fmaf(pv, as[f], s0);
    s1 = fmaf(pv, at[f], s1);
  }
  ssrc[idx] = s0;
  stgt[idx] = s1;
}

// ---------------------------------------------------------------------------
// Softmax row stats: one wave per (h, i); lanes sweep j.
// Sweep 1: row max; sweep 2: sum exp(e - max). Exact (matches jax softmax).
// ---------------------------------------------------------------------------
__global__ void __launch_bounds__(256)
k_softmax_stats(const float* __restrict__ adj, const float* __restrict__ ssrc,
                const float* __restrict__ stgt,
                float* __restrict__ rmax, float* __restrict__ rden, int pairs)
{
  int wid = (blockIdx.x * 256 + threadIdx.x) >> 5;   // uniform per wave
  int lane = threadIdx.x & 31;
  if (wid >= pairs) return;
  int h = wid >> 12;                                 // wid / GN
  int i = wid & (GN - 1);
  float ss = ssrc[wid];                              // [H,N] layout
  const float* st = stgt + (size_t)h * GN;
  const float* ar = adj + (size_t)i * GN;

  float m = -1e30f;
  for (int j = lane; j < GN; j += 32) {
    float e = lrelu02(ss + st[j]);
    e = ar[j] > 0.f ? e : -1e9f;
    m = fmaxf(m, e);
  }
#pragma unroll
  for (int o = 16; o > 0; o >>= 1) m = fmaxf(m, __shfl_xor(m, o, 32));

  float d = 0.f;
  for (int j = lane; j < GN; j += 32) {
    float e = lrelu02(ss + st[j]);
    e = ar[j] > 0.f ? e : -1e9f;
    d += __expf(e - m);
  }
#pragma unroll
  for (int o = 16; o > 0; o >>= 1) d += __shfl_xor(d, o, 32);

  if (lane == 0) { rmax[wid] = m; rden[wid] = d; }
}

// ---------------------------------------------------------------------------
// Layer-1 aggregation via V_WMMA_F32_16X16X4_F32.
// Block = 128 threads = 4 waves; wave w handles head w for the block's 16-row
// tile, over all 64 output features (4 C/D accumulators), sweeping j in 4s.
// A-operand layout (16x4 f32): lanes 0-15 carry K=0,1 ; lanes 16-31 K=2,3,
// so each lane evaluates exactly its own two exp(e-m) weights.
// Writeback fuses 1/denominator, skip connection, bias and ELU.
// ---------------------------------------------------------------------------
__global__ void __launch_bounds__(128)
k_l1_aggregate(const float* __restrict__ adj, const float* __restrict__ proj,
               const float* __restrict__ skipp, const float* __restrict__ bias,
               const float* __restrict__ ssrc, const float* __restrict__ stgt,
               const float* __restrict__ rmax, const float* __restrict__ rden,
               float* __restrict__ out)   // h buffer [N, 256]
{
  const int h    = threadIdx.x >> 5;     // wave id == head
  const int lane = threadIdx.x & 31;
  const int half = lane >> 4;
  const int l    = lane & 15;
  const int i0   = blockIdx.x * 16;
  const int i    = i0 + l;               // A-matrix row owned by this lane
  const int jb   = half * 2;             // K offset for this lane half
  const int cb   = h * GF1 + l;          // output column base (N-index = l)

  const float ss = ssrc[h * GN + i];
  const float mi = rmax[h * GN + i];
  const float* st   = stgt + (size_t)h * GN;
  const float* arow = adj + (size_t)i * GN;

  v8f acc0 = {}, acc1 = {}, acc2 = {}, acc3 = {};

  for (int j = 0; j < GN; j += 4) {
    // attention weights for A[i, j+jb], A[i, j+jb+1]
    float e0 = lrelu02(ss + st[j + jb]);
    float e1 = lrelu02(ss + st[j + jb + 1]);
    float a0 = arow[j + jb];
    float a1 = arow[j + jb + 1];
    e0 = a0 > 0.f ? e0 : -1e9f;
    e1 = a1 > 0.f ? e1 : -1e9f;
    v2f A;
    A.x = __expf(e0 - mi);
    A.y = __expf(e1 - mi);

    // B: proj rows (j+jb, j+jb+1), columns cb + {0,16,32,48}
    const float* p0 = proj + (size_t)(j + jb) * GC1 + cb;
    v2f B0 = { p0[0],  p0[GC1 + 0]  };
    v2f B1 = { p0[16], p0[GC1 + 16] };
    v2f B2 = { p0[32], p0[GC1 + 32] };
    v2f B3 = { p0[48], p0[GC1 + 48] };

    acc0 = __builtin_amdgcn_wmma_f32_16x16x4_f32(false, A, false, B0, (short)0, acc0, false, false);
    acc1 = __builtin_amdgcn_wmma_f32_16x16x4_f32(false, A, false, B1, (short)0, acc1, false, false);
    acc2 = __builtin_amdgcn_wmma_f32_16x16x4_f32(false, A, false, B2, (short)0, acc2, false, false);
    acc3 = __builtin_amdgcn_wmma_f32_16x16x4_f32(false, A, false, B3, (short)0, acc3, false, false);
  }

  // D layout: VGPR r -> row i0+r (lanes 0-15) / i0+r+8 (lanes 16-31), col = l
#pragma unroll
  for (int r = 0; r < 8; ++r) {
    int row = i0 + r + 8 * half;
    float invd = 1.f / rden[h * GN + row];
    float v0 = acc0[r] * invd + skipp[row * GC1 + cb +  0] + bias[cb +  0];
    float v1 = acc1[r] * invd + skipp[row * GC1 + cb + 16] + bias[cb + 16];
    float v2 = acc2[r] * invd + skipp[row * GC1 + cb + 32] + bias[cb + 32];
    float v3 = acc3[r] * invd + skipp[row * GC1 + cb + 48] + bias[cb + 48];
    // ELU
    v0 = v0 > 0.f ? v0 : __expf(v0) - 1.f;
    v1 = v1 > 0.f ? v1 : __expf(v1) - 1.f;
    v2 = v2 > 0.f ? v2 : __expf(v2) - 1.f;
    v3 = v3 > 0.f ? v3 : __expf(v3) - 1.f;
    out[row * GC1 + cb +  0] = v0;
    out[row * GC1 + cb + 16] = v1;
    out[row * GC1 + cb + 32] = v2;
    out[row * GC1 + cb + 48] = v3;
  }
}

// ---------------------------------------------------------------------------
// Layer-2 projections: proj2 = h@w2, skipp2 = h@skip2  (K=256)
// one thread per (n, c) of [N, 32]
// ---------------------------------------------------------------------------
__global__ void __launch_bounds__(256)
k_l2_proj(const float* __restrict__ hbuf, const float* __restrict__ w2,
          const float* __restrict__ skip2,
          float* __restrict__ proj, float* __restrict__ skipp)
{
  int idx = blockIdx.x * 256 + threadIdx.x;      // < N*32
  int n = idx >> 5, c = idx & (GF2 - 1);
  const float* hr = hbuf + (size_t)n * GC1;
  float p = 0.f, s = 0.f;
#pragma unroll 8
  for (int k = 0; k < GC1; ++k) {
    float hv = hr[k];
    p = fmaf(hv, w2[k * GF2 + c], p);
    s = fmaf(hv, skip2[k * GF2 + c], s);
  }
  proj[idx] = p;
  skipp[idx] = s;
}

__global__ void __launch_bounds__(256)
k_l2_scores(const float* __restrict__ proj, const float* __restrict__ asrc,
            const float* __restrict__ atgt,
            float* __restrict__ ssrc, float* __restrict__ stgt)
{
  int n = blockIdx.x * 256 + threadIdx.x;        // < N
  const float* pr = proj + n * GF2;
  float s0 = 0.f, s1 = 0.f;
#pragma unroll
  for (int f = 0; f < GF2; ++f) {
    float pv = pr[f];
    s0 = fmaf(pv, asrc[f], s0);
    s1 = fmaf(pv, atgt[f], s1);
  }
  ssrc[n] = s0;
  stgt[n] = s1;
}

// ---------------------------------------------------------------------------
// Layer-2 aggregation (H=1, F=32): one wave per 16-row tile, 2 accumulators.
// Writes node_emb straight into d_out (no activation; mean over 1 head = id).
// ---------------------------------------------------------------------------
__global__ void __launch_bounds__(32)
k_l2_aggregate(const float* __restrict__ adj, const float* __restrict__ proj,
               const float* __restrict__ skipp, const float* __restrict__ bias,
               const float* __restrict__ ssrc, const float* __restrict__ stgt,
               const float* __restrict__ rmax, const float* __restrict__ rden,
               float* __restrict__ out)   // node_emb [N, 32] (d_out)
{
  const int lane = threadIdx.x;
  const int half = lane >> 4;
  const int l    = lane & 15;
  const int i0   = blockIdx.x * 16;
  const int i    = i0 + l;
  const int jb   = half * 2;

  const float ss = ssrc[i];
  const float mi = rmax[i];
  const float* arow = adj + (size_t)i * GN;

  v8f acc0 = {}, acc1 = {};

  for (int j = 0; j < GN; j += 4) {
    float e0 = lrelu02(ss + stgt[j + jb]);
    float e1 = lrelu02(ss + stgt[j + jb + 1]);
    float a0 = arow[j + jb];
    float a1 = arow[j + jb + 1];
    e0 = a0 > 0.f ? e0 : -1e9f;
    e1 = a1 > 0.f ? e1 : -1e9f;
    v2f A;
    A.x = __expf(e0 - mi);
    A.y = __expf(e1 - mi);

    const float* p0 = proj + (size_t)(j + jb) * GF2 + l;
    v2f B0 = { p0[0],  p0[GF2 + 0]  };
    v2f B1 = { p0[16], p0[GF2 + 16] };

    acc0 = __builtin_amdgcn_wmma_f32_16x16x4_f32(false, A, false, B0, (short)0, acc0, false, false);
    acc1 = __builtin_amdgcn_wmma_f32_16x16x4_f32(false, A, false, B1, (short)0, acc1, false, false);
  }

#pragma unroll
  for (int r = 0; r < 8; ++r) {
    int row = i0 + r + 8 * half;
    float invd = 1.f / rden[row];
    out[row * GF2 + l]      = acc0[r] * invd + skipp[row * GF2 + l]      + bias[l];
    out[row * GF2 + l + 16] = acc1[r] * invd + skipp[row * GF2 + l + 16] + bias[l + 16];
  }
}

// ---------------------------------------------------------------------------
// Graph pooling + MLP + LayerNorm, single 256-thread block.
// scores -> softmax over nodes -> weighted sum -> relu MLP -> layernorm.
// ---------------------------------------------------------------------------
__global__ void __launch_bounds__(256)
k_pool(const float* __restrict__ emb,       // node_emb [N,32] (d_out)
       const float* __restrict__ aw1, const float* __restrict__ ab1,
       const float* __restrict__ aw2, const float* __restrict__ ab2,
       const float* __restrict__ lg, const float* __restrict__ lb,
       float* __restrict__ out)             // graph_emb [32]
{
  __shared__ float sc[GN];          // 16 KB of the 320 KB LDS
  __shared__ float red[8];
  __shared__ float gsum[GF2];
  __shared__ float hid[GF1];
  const int tid = threadIdx.x;
  const int lane = tid & 31;
  const int wv = tid >> 5;

  // scores + block max
  float lmax = -1e30f;
  for (int n = tid; n < GN; n += 256) {
    const float* v = emb + n * GF2;
    float s = 0.f;
#pragma unroll
    for (int c = 0; c < GF2; ++c) { float t = v[c]; s = fmaf(t, t, s); }
    sc[n] = s;
    lmax = fmaxf(lmax, s);
  }
#pragma unroll
  for (int o = 16; o > 0; o >>= 1) lmax = fmaxf(lmax, __shfl_xor(lmax, o, 32));
  if (lane == 0) red[wv] = lmax;
  __syncthreads();
  float M = red[0];
#pragma unroll
  for (int w = 1; w < 8; ++w) M = fmaxf(M, red[w]);
  __syncthreads();

  // sum of exp
  float lsum = 0.f;
  for (int n = tid; n < GN; n += 256) lsum += __expf(sc[n] - M);
#pragma unroll
  for (int o = 16; o > 0; o >>= 1) lsum += __shfl_xor(lsum, o, 32);
  if (lane == 0) red[wv] = lsum;
  if (tid < GF2) gsum[tid] = 0.f;
  __syncthreads();
  float D = 0.f;
#pragma unroll
  for (int w = 0; w < 8; ++w) D += red[w];

  // weighted sum of embeddings (unnormalized; divide by D later)
  float g[GF2];
#pragma unroll
  for (int c = 0; c < GF2; ++c) g[c] = 0.f;
  for (int n = tid; n < GN; n += 256) {
    float w = __expf(sc[n] - M);
    const float* v = emb + n * GF2;
#pragma unroll
    for (int c = 0; c < GF2; ++c) g[c] = fmaf(w, v[c], g[c]);
  }
#pragma unroll
  for (int c = 0; c < GF2; ++c) atomicAdd(&gsum[c], g[c]);   // ds_add_f32
  __syncthreads();

  // hid = relu(g @ agg_w1 + agg_b1)   [64]
  if (tid < GF1) {
    float a = ab1[tid];
#pragma unroll
    for (int c = 0; c < GF2; ++c) a = fmaf(gsum[c] / D, aw1[c * GF1 + tid], a);
    hid[tid] = a > 0.f ? a : 0.f;
  }
  __syncthreads();

  // g2 = hid @ agg_w2 + agg_b2, then LayerNorm across 32 lanes of wave 0
  if (tid < GF2) {
    float a = ab2[tid];
#pragma unroll
    for (int f = 0; f < GF1; ++f) a = fmaf(hid[f], aw2[f * GF2 + tid], a);
    float mu = a;
#pragma unroll
    for (int o = 16; o > 0; o >>= 1) mu += __shfl_xor(mu, o, 32);
    mu *= (1.f / 32.f);
    float d = a - mu;
    float var = d * d;
#pragma unroll
    for (int o = 16; o > 0; o >>= 1) var += __shfl_xor(var, o, 32);
    var *= (1.f / 32.f);
    out[tid] = d * rsqrtf(var + 1e-5f) * lg[tid] + lb[tid];
  }
}

// ---------------------------------------------------------------------------
extern "C" void kernel_launch(void* const* d_in, const int* in_sizes, int n_in,
                              void* d_out, int out_size, void* d_ws, size_t ws_size,
                              hipStream_t stream) {
  (void)in_sizes; (void)n_in; (void)out_size; (void)ws_size;

  const float* x     = (const float*)d_in[0];
  const float* adj   = (const float*)d_in[1];
  const float* w1    = (const float*)d_in[2];
  const float* as1   = (const float*)d_in[3];
  const float* at1   = (const float*)d_in[4];
  const float* sk1   = (const float*)d_in[5];
  const float* b1    = (const float*)d_in[6];
  const float* w2    = (const float*)d_in[7];
  const float* as2   = (const float*)d_in[8];
  const float* at2   = (const float*)d_in[9];
  const float* sk2   = (const float*)d_in[10];
  const float* b2    = (const float*)d_in[11];
  const float* aw1   = (const float*)d_in[12];
  const float* ab1   = (const float*)d_in[13];
  const float* aw2   = (const float*)d_in[14];
  const float* ab2   = (const float*)d_in[15];
  const float* lng   = (const float*)d_in[16];
  const float* lnb   = (const float*)d_in[17];
  float* out = (float*)d_out;

  // workspace carve-up (~13.5 MB of fp32)
  float* W = (float*)d_ws;
  float* proj1  = W;                   W += (size_t)GN * GC1;
  float* skipp1 = W;                   W += (size_t)GN * GC1;
  float* ssrc1  = W;                   W += (size_t)GH * GN;
  float* stgt1  = W;                   W += (size_t)GH * GN;
  float* m1     = W;                   W += (size_t)GH * GN;
  float* dn1    = W;                   W += (size_t)GH * GN;
  float* hbuf   = W;                   W += (size_t)GN * GC1;
  float* proj2  = W;                   W += (size_t)GN * GF2;
  float* skipp2 = W;                   W += (size_t)GN * GF2;
  float* ssrc2  = W;                   W += (size_t)GN;
  float* stgt2  = W;                   W += (size_t)GN;
  float* m2     = W;                   W += (size_t)GN;
  float* dn2    = W;                   W += (size_t)GN;

  // ---- layer 1 ----
  k_l1_proj<<<GN, 256, 0, stream>>>(x, w1, sk1, proj1, skipp1);
  k_l1_scores<<<(GH * GN) / 256, 256, 0, stream>>>(proj1, as1, at1, ssrc1, stgt1);
  k_softmax_stats<<<(GH * GN * 32) / 256, 256, 0, stream>>>(adj, ssrc1, stgt1, m1, dn1, GH * GN);
  k_l1_aggregate<<<GN / 16, 128, 0, stream>>>(adj, proj1, skipp1, b1,
                                              ssrc1, stgt1, m1, dn1, hbuf);
  // ---- layer 2 ----
  k_l2_proj<<<(GN * GF2) / 256, 256, 0, stream>>>(hbuf, w2, sk2, proj2, skipp2);
  k_l2_scores<<<GN / 256, 256, 0, stream>>>(proj2, as2, at2, ssrc2, stgt2);
  k_softmax_stats<<<(GN * 32) / 256, 256, 0, stream>>>(adj, ssrc2, stgt2, m2, dn2, GN);
  k_l2_aggregate<<<GN / 16, 32, 0, stream>>>(adj, proj2, skipp2, b2,
                                             ssrc2, stgt2, m2, dn2, out);
  // ---- pooling head ----
  k_pool<<<1, 256, 0, stream>>>(out, aw1, ab1, aw2, ab2, lng, lnb, out + (size_t)GN * GF2);
}